// CPC_Loss_63127429316797
// MI455X (gfx1250) — compile-verified
//
#include <hip/hip_runtime.h>

// CPC loss, MI455X (gfx1250, wave32).
// Reduced form: loss = (1/K) * sum_i [ -S_i / (B*(T-i)) ],
//   S_i = sum_{b, t<T-i, t<seq_len[b]} dot(mctx[b,t,:,i-1], base[b,t+i,:]-negsum[b,:])
// HBM-bound: ~268MB mapped_ctx stream -> ~13us at 23.3 TB/s.
// MAC engine: V_WMMA_F32_16X16X4_F32, dot products on the 16x16 diagonal (f32 exact).

#define Bq 64
#define Tq 1024
#define Eq 128
#define Kq 8
#define Nq 64

typedef __attribute__((ext_vector_type(2))) float v2f;
typedef __attribute__((ext_vector_type(8))) float v8f;

#define WAVES 4
#define TT (WAVES * 16)         // 64 t-rows per block (one 16-row tile per wave)
#define ROWS (TT + Kq - 1)      // 71 staged diff rows (t+1 .. t+TT+K-1)
#define RPAD 132                // row pad: stride%64==4 -> conflict-free ds_load_b64

// ---------------- kernel 1: negsum[b][e] = sum_n base[neg_ids[b,n]][e] -------
__global__ void negsum_kernel(const float* __restrict__ base,
                              const int* __restrict__ neg_ids,
                              float* __restrict__ negsum) {
  const int b = blockIdx.x;
  const int e = threadIdx.x;          // 128 threads
  float s = 0.f;
  #pragma unroll 4
  for (int n = 0; n < Nq; ++n) {
    const int id = neg_ids[b * Nq + n];          // flat row into (B*T, E)
    s += base[(size_t)id * Eq + e];
  }
  negsum[b * Eq + e] = s;
}

// ---------------- kernel 2: main WMMA reduction ------------------------------
__global__ __launch_bounds__(WAVES * 32) void
cpc_main_kernel(const float* __restrict__ base,
                const float* __restrict__ mctx,
                const int*   __restrict__ seq_lens,
                const float* __restrict__ negsum,
                float* __restrict__ partials) {
  __shared__ float sDiff[ROWS * RPAD];   // diff tile: base[t0blk+1+r][e]-negsum[e]
  __shared__ float sRed[WAVES * Kq];

  const int b      = blockIdx.y;
  const int tblk   = blockIdx.x;
  const int t0blk  = tblk * TT;
  const int tid    = threadIdx.x;
  const int w      = tid >> 5;
  const int lane   = tid & 31;
  const int seqLen = seq_lens[b];

  // ---- stage DIFF into LDS (float4 loads, rows past T-1 staged as 0-negsum;
  //      harmless: the matching A rows are zeroed so only finite-ness matters)
  {
    const float4* nsum4 = (const float4*)(negsum + (size_t)b * Eq);
    for (int f = tid; f < ROWS * (Eq / 4); f += WAVES * 32) {
      const int r    = f >> 5;                 // Eq/4 == 32 chunks per row
      const int e4   = f & 31;
      const int trow = t0blk + 1 + r;
      float4 v = make_float4(0.f, 0.f, 0.f, 0.f);
      if (trow < Tq)
        v = ((const float4*)(base + (size_t)(b * Tq + trow) * Eq))[e4];
      const float4 ns = nsum4[e4];
      const int o = r * RPAD + e4 * 4;
      sDiff[o + 0] = v.x - ns.x;
      sDiff[o + 1] = v.y - ns.y;
      sDiff[o + 2] = v.z - ns.z;
      sDiff[o + 3] = v.w - ns.w;
    }
  }
  __syncthreads();

  // ---- per-wave 16-row t tile, all K=8 steps in 8 f32 WMMA accumulators ----
  const int t0   = t0blk + w * 16;   // uniform per wave
  const int m    = lane & 15;        // A row / B col index this lane carries
  const int half = lane >> 4;        // selects K=0..1 (half=0) vs K=2..3 (half=1)
  const int t_a  = t0 + m;

  v8f c[Kq] = {};                    // 8 zeroed 16x16 f32 accumulators

  if (t0 < seqLen) {                 // wave-uniform skip of fully masked tiles
    int vk = 0;                      // #valid steps for this lane's row
    if (t_a < seqLen) {
      vk = (Tq - 1) - t_a;
      vk = vk > Kq ? Kq : (vk < 0 ? 0 : vk);
    }
    for (int e0 = 0; e0 < Eq; e0 += 4) {
      const int e_a = e0 + 2 * half;
      float m0[Kq], m1[Kq];
      if (vk > 0) {
        // 64 contiguous bytes per lane; lane L + lane L+16 = one full 128B line
        const float4* mp =
            (const float4*)(mctx + (size_t)((b * Tq + t_a) * Eq + e_a) * Kq);
        const float4 A0 = mp[0], A1 = mp[1];   // mctx[t_a][e_a  ][k=0..7]
        const float4 A2 = mp[2], A3 = mp[3];   // mctx[t_a][e_a+1][k=0..7]
        m0[0]=A0.x; m0[1]=A0.y; m0[2]=A0.z; m0[3]=A0.w;
        m0[4]=A1.x; m0[5]=A1.y; m0[6]=A1.z; m0[7]=A1.w;
        m1[0]=A2.x; m1[1]=A2.y; m1[2]=A2.z; m1[3]=A2.w;
        m1[4]=A3.x; m1[5]=A3.y; m1[6]=A3.z; m1[7]=A3.w;
        #pragma unroll
        for (int kk = 0; kk < Kq; ++kk)
          if (kk >= vk) { m0[kk] = 0.f; m1[kk] = 0.f; }
      } else {
        #pragma unroll
        for (int kk = 0; kk < Kq; ++kk) { m0[kk] = 0.f; m1[kk] = 0.f; }
      }
      #pragma unroll
      for (int kk = 0; kk < Kq; ++kk) {        // step i = kk+1
        // B operand: diff_i row rr = (t0+m+i) - (t0blk+1) = w*16 + m + kk
        const int rr = (w * 16 + m) + kk;
        const v2f bv = *(const v2f*)&sDiff[rr * RPAD + e_a];
        v2f av; av.x = m0[kk]; av.y = m1[kk];
        c[kk] = __builtin_amdgcn_wmma_f32_16x16x4_f32(
                    false, av, false, bv, (short)0, c[kk], false, false);
      }
    }
  }

  // ---- diagonal extract (D[m][m]) + wave32 butterfly reduce ----
  // diag lives at: m=0..7 -> lane m, vgpr m ; m=8..15 -> lane m+16, vgpr m-8
  const int sel = (lane < 8) ? lane : (lane >= 24 ? lane - 24 : -1);
  #pragma unroll
  for (int kk = 0; kk < Kq; ++kk) {
    float x = 0.f;
    #pragma unroll
    for (int j = 0; j < 8; ++j) x = (sel == j) ? c[kk][j] : x;
    #pragma unroll
    for (int off = 16; off > 0; off >>= 1) x += __shfl_xor(x, off, 32);
    if (lane == 0) sRed[w * Kq + kk] = x;
  }
  __syncthreads();
  if (tid < Kq) {
    float p = 0.f;
    #pragma unroll
    for (int ww = 0; ww < WAVES; ++ww) p += sRed[ww * Kq + tid];
    const int blockId = b * gridDim.x + tblk;
    partials[(size_t)blockId * Kq + tid] = p;   // deterministic two-stage reduce
  }
}

// ---------------- kernel 3: deterministic finalize ---------------------------
__global__ void finalize_kernel(const float* __restrict__ partials,
                                int nblocks, float* __restrict__ out) {
  __shared__ float step[Kq];
  const int kk = threadIdx.x;
  if (kk < Kq) {
    float s = 0.f;
    for (int p = 0; p < nblocks; ++p) s += partials[(size_t)p * Kq + kk];
    step[kk] = -s / (float)(Bq * (Tq - (kk + 1)));
  }
  __syncthreads();
  if (kk == 0) {
    float t = 0.f;
    #pragma unroll
    for (int j = 0; j < Kq; ++j) t += step[j];
    out[0] = t / (float)Kq;
  }
}

extern "C" void kernel_launch(void* const* d_in, const int* in_sizes, int n_in,
                              void* d_out, int out_size, void* d_ws, size_t ws_size,
                              hipStream_t stream) {
  const float* base     = (const float*)d_in[0];   // (B,T,E) f32
  const float* mctx     = (const float*)d_in[1];   // (B,T,E,K) f32
  const int*   seq_lens = (const int*)d_in[2];     // (B,)
  const int*   neg_ids  = (const int*)d_in[3];     // (B,N)

  float* negsum   = (float*)d_ws;                  // B*E floats   (32 KB)
  float* partials = negsum + (size_t)Bq * Eq;      // nblocks*K    (32 KB)
  float* out      = (float*)d_out;

  negsum_kernel<<<dim3(Bq), dim3(Eq), 0, stream>>>(base, neg_ids, negsum);

  dim3 grid(Tq / TT, Bq);                          // 16 x 64 = 1024 blocks
  cpc_main_kernel<<<grid, dim3(WAVES * 32), 0, stream>>>(
      base, mctx, seq_lens, negsum, partials);

  const int nblocks = (Tq / TT) * Bq;
  finalize_kernel<<<1, 64, 0, stream>>>(partials, nblocks, out);
}